// Latent_Space_56719338111582
// MI455X (gfx1250) — compile-verified
//
#include <hip/hip_runtime.h>

typedef __attribute__((ext_vector_type(2))) float v2f;
typedef __attribute__((ext_vector_type(8))) float v8f;

#define NUM_EMB   512
#define N_ELEM    (16 * 128 * 128)      // 262144 f32 elements
#define TILE      16
#define NTILES_X  (N_ELEM / TILE)       // 16384 input tiles
#define CB_TILES  (NUM_EMB / 16)        // 32 codebook tiles
#define BLOCKS    256
#define THREADS   256

// VQ nearest-codebook search via the GEMM trick:
//   score(e, x) = e^2 - 2*x*e   (argmin_e score == argmin_e (x-e)^2)
// D = A x B with A(16x4) = [e, e^2, 0, 0] rows, B(4x16) = [-2x ; 1 ; 0 ; 0]
// -> one V_WMMA_F32_16X16X4_F32 scores a 16-codebook x 16-input tile.
__global__ __launch_bounds__(THREADS) void vq_wmma_kernel(
    const float* __restrict__ x, const float* __restrict__ cb,
    float* __restrict__ out, float* __restrict__ blockSums)
{
    __shared__ float lds_cb[NUM_EMB];
    __shared__ float lds_red[THREADS];

    for (int i = threadIdx.x; i < NUM_EMB; i += THREADS) lds_cb[i] = cb[i];
    __syncthreads();

    const int  lane = threadIdx.x & 31;
    const int  wave = threadIdx.x >> 5;
    const int  gwave  = blockIdx.x * (THREADS / 32) + wave;
    const int  nwaves = gridDim.x * (THREADS / 32);
    const bool lo   = lane < 16;        // lanes 0-15: K=0,1 rows / M=0..7 half
    const int  col  = lane & 15;        // input element within tile (N index)
    const int  hi8  = lo ? 0 : 8;       // D rows M=r (+8 for upper lane half)

    float acc = 0.0f;                   // per-thread (q-x)^2 partial sum

    for (int tile = gwave; tile < NTILES_X; tile += nwaves) {
        const int tbase = tile * TILE;
        const float xv = x[tbase + col];

        // B (4x16): VGPR0 lanes0-15 = K=0 row (-2x), VGPR1 lanes0-15 = K=1 row (1)
        v2f B;
        B.x = lo ? (-2.0f * xv) : 0.0f;
        B.y = lo ? 1.0f : 0.0f;

        float best = 3.402823466e38f;
        int   bidx = 0;

        for (int t = 0; t < CB_TILES; ++t) {
            const float e = lds_cb[t * 16 + col];
            // A (16x4): VGPR0 lanes0-15 = K=0 col (e), VGPR1 lanes0-15 = K=1 col (e^2)
            v2f A;
            A.x = lo ? e : 0.0f;
            A.y = lo ? (e * e) : 0.0f;

            v8f C = {0.f, 0.f, 0.f, 0.f, 0.f, 0.f, 0.f, 0.f};
            // D[r] = score for codebook entry (t*16 + r + hi8), input (tbase+col)
            v8f D = __builtin_amdgcn_wmma_f32_16x16x4_f32(
                false, A, false, B, (short)0, C, false, false);

            #pragma unroll
            for (int r = 0; r < 8; ++r) {
                const int   idx = t * 16 + r + hi8;
                const float s   = D[r];
                if (s < best) { best = s; bidx = idx; }  // strict < keeps lowest idx
            }
        }

        // Merge the two lane halves (same x, disjoint codebook rows).
        const float obest = __shfl_xor(best, 16, 32);
        const int   oidx  = __shfl_xor(bidx, 16, 32);
        if (obest < best || (obest == best && oidx < bidx)) { best = obest; bidx = oidx; }

        const float q = lds_cb[bidx];
        if (lo) {   // each x counted exactly once
            out[tbase + col] = q;
            const float d = q - xv;
            acc += d * d;
        }
    }

    // Deterministic in-order block reduction of the loss partials.
    lds_red[threadIdx.x] = acc;
    __syncthreads();
    if (threadIdx.x == 0) {
        float s = 0.0f;
        for (int i = 0; i < THREADS; ++i) s += lds_red[i];
        blockSums[blockIdx.x] = s;
    }
}

__global__ void vq_finalize(const float* __restrict__ blockSums, int nblocks,
                            float* __restrict__ loss)
{
    if (blockIdx.x == 0 && threadIdx.x == 0) {
        float s = 0.0f;
        for (int i = 0; i < nblocks; ++i) s += blockSums[i];
        // loss1 + 0.25*loss2, and loss1 == loss2 == mean((q-x)^2) when C==1
        *loss = 1.25f * s / (float)N_ELEM;
    }
}

extern "C" void kernel_launch(void* const* d_in, const int* in_sizes, int n_in,
                              void* d_out, int out_size, void* d_ws, size_t ws_size,
                              hipStream_t stream) {
    const float* x   = (const float*)d_in[0];   // pre_quantized: 262144 f32
    const float* cb  = (const float*)d_in[1];   // emb_weight: 512 f32
    float* out       = (float*)d_out;           // [0..N) quantized, [N] loss
    float* blockSums = (float*)d_ws;            // BLOCKS f32 partials

    vq_wmma_kernel<<<BLOCKS, THREADS, 0, stream>>>(x, cb, out, blockSums);
    vq_finalize<<<1, 32, 0, stream>>>(blockSums, BLOCKS, out + N_ELEM);
}